// Net_2585570312759
// MI455X (gfx1250) — compile-verified
//
#include <hip/hip_runtime.h>

// Problem constants (match reference file)
#define NN 100000
#define RR 50
#define BB 30
#define HH 16
#define CC 8
#define EE 1600000

typedef __attribute__((ext_vector_type(2))) float v2f;
typedef __attribute__((ext_vector_type(8))) float v8f;

// Workspace layout (floats):
//   [0, N*R)              cnt   (per (dst,rel) edge counts)
//   [N*R, N*R+N*H)        h     (layer-1 node features)
//   [N*R+N*H, +R*H*C)     w2t   (comp2-combined layer-2 weights, TRANSPOSED [r][c][h])
#define CNT_OFF 0
#define H_OFF   (NN * RR)
#define W2_OFF  (NN * RR + NN * HH)
#define ZERO_TOTAL (NN * RR + NN * HH)   // divisible by 4

__global__ void k_zero(float4* ws4) {
  int i = blockIdx.x * blockDim.x + threadIdx.x;
  if (i < ZERO_TOTAL / 4) ws4[i] = make_float4(0.f, 0.f, 0.f, 0.f);
}

// cnt[dst*R + rel] += 1
__global__ void k_count(const int* __restrict__ ei, const int* __restrict__ et,
                        float* __restrict__ cnt) {
  int e = blockIdx.x * blockDim.x + threadIdx.x;
  if (e >= EE) return;
  int dst = ei[EE + e];
  int rel = et[e];
  atomicAdd(&cnt[dst * RR + rel], 1.0f);
}

// w2t[r, c, h] = sum_b comp2[r,b] * basis2[b,h,c]   (50*8*16 = 6400 outputs, transposed)
__global__ void k_w2(const float* __restrict__ comp2, const float* __restrict__ basis2,
                     float* __restrict__ w2t) {
  int i = blockIdx.x * blockDim.x + threadIdx.x;
  if (i >= RR * HH * CC) return;
  int r = i / (HH * CC);
  int c = (i / HH) % CC;
  int h = i % HH;
  float acc = 0.0f;
  for (int b = 0; b < BB; ++b)
    acc = fmaf(comp2[r * BB + b], basis2[b * HH * CC + h * CC + c], acc);
  w2t[i] = acc;   // layout [r][c][h]
}

// Layer-1 edge scatter: 16 lanes per edge, lane = h index.
// h[dst,hh] += inv * sum_b comp1[rel,b] * basis1[b,src,hh]
__global__ void k_layer1(const int* __restrict__ ei, const int* __restrict__ et,
                         const float* __restrict__ basis1, const float* __restrict__ comp1,
                         const float* __restrict__ cnt, float* __restrict__ hbuf) {
  __shared__ float s_comp[RR * BB];   // 6 KB of 320 KB LDS
  int tid = threadIdx.x;
  for (int i = tid; i < RR * BB; i += 256) s_comp[i] = comp1[i];
  __syncthreads();

  int e = blockIdx.x * 16 + (tid >> 4);
  int hh = tid & 15;
  if (e >= EE) return;
  int src = ei[e];
  int dst = ei[EE + e];
  int rel = et[e];

  const float* cr = &s_comp[rel * BB];
  long base = (long)src * HH + hh;
  float acc = 0.0f;
  #pragma unroll
  for (int b = 0; b < BB; ++b)
    acc = fmaf(cr[b], basis1[(long)b * (NN * HH) + base], acc);   // 64B coalesced, L2-resident

  float inv = 1.0f / fmaxf(cnt[dst * RR + rel], 1.0f);
  atomicAdd(&hbuf[dst * HH + hh], acc * inv);
}

// h += root1 + bias1  (float4-vectorized; N*H divisible by 4)
__global__ void k_hfinal(const float4* __restrict__ root1_4, const float* __restrict__ bias1,
                         float4* __restrict__ hbuf4) {
  int i = blockIdx.x * blockDim.x + threadIdx.x;
  if (i >= NN * HH / 4) return;
  const float4* b4 = (const float4*)bias1;   // bias1 has 16 floats = 4 float4s
  float4 hv = hbuf4[i];
  float4 rv = root1_4[i];
  float4 bv = b4[i & 3];
  hv.x += rv.x + bv.x;
  hv.y += rv.y + bv.y;
  hv.z += rv.z + bv.z;
  hv.w += rv.w + bv.w;
  hbuf4[i] = hv;
}

// out[n,c] = (h @ root2)[n,c] + bias2[c] via V_WMMA_F32_16X16X4_F32.
// 2 waves/block, one 16-node tile per wave; K=16 as 4 steps of K=4;
// C padded 8->16 with a multiplicative mask (branch-free, EXEC stays all-1s).
__global__ void k_root_wmma(const float* __restrict__ hbuf, const float* __restrict__ root2,
                            const float* __restrict__ bias2, float* __restrict__ out) {
  int lane = threadIdx.x & 31;                       // 0..31 within wave
  int wave = threadIdx.x >> 5;                       // 0..1
  int half = lane >> 4;
  int col  = lane & 15;                              // M index for A, N index for B/D
  int base = (blockIdx.x * 2 + wave) * 16;           // node tile base

  float mask = (col < CC) ? 1.0f : 0.0f;
  int   bcol = col & 7;                              // clamped B column (masked to 0 later)

  v8f c = {};
  #pragma unroll
  for (int kk = 0; kk < 4; ++kk) {
    // A 16x4 f32: VGPR0 holds K = 2*half, VGPR1 holds K = 2*half+1 (ISA 7.12.2)
    int k0 = 4 * kk + 2 * half;
    v2f a, b;
    a.x = hbuf[(base + col) * HH + k0];
    a.y = hbuf[(base + col) * HH + k0 + 1];
    // B 4x16 f32: row K striped across lanes, same half split; cols 8..15 are zero
    b.x = root2[k0 * CC + bcol] * mask;
    b.y = root2[(k0 + 1) * CC + bcol] * mask;
    c = __builtin_amdgcn_wmma_f32_16x16x4_f32(
        /*neg_a=*/false, a, /*neg_b=*/false, b,
        /*c_mod=*/(short)0, c, /*reuse_a=*/false, /*reuse_b=*/false);
  }

  // D 16x16 f32: VGPR i -> row (i + 8*half), col = lane&15. Keep first 8 cols.
  if (col < CC) {
    float bv = bias2[col];
    #pragma unroll
    for (int i = 0; i < 8; ++i) {
      int node = base + i + 8 * half;
      out[node * CC + col] = c[i] + bv;
    }
  }
}

// Layer-2 edge scatter: 8 lanes per edge, lane = class index.
// out[dst,c] += inv * sum_h h[src,h] * w2[rel,h,c]
// w2 stored transposed [r][c][h] so both h[src] and w2 reads are 4x b128 vector loads.
__global__ void k_layer2(const int* __restrict__ ei, const int* __restrict__ et,
                         const float* __restrict__ hbuf, const float* __restrict__ w2t,
                         const float* __restrict__ cnt, float* __restrict__ out) {
  __shared__ float s_w2[RR * HH * CC];   // 25.6 KB of 320 KB LDS
  int tid = threadIdx.x;
  for (int i = tid; i < RR * HH * CC; i += 256) s_w2[i] = w2t[i];
  __syncthreads();

  int e = blockIdx.x * 32 + (tid >> 3);
  int cc = tid & 7;
  if (e >= EE) return;
  int src = ei[e];
  int dst = ei[EE + e];
  int rel = et[e];

  const float4* hn4 = (const float4*)&hbuf[src * HH];               // 64B-aligned
  const float4* wr4 = (const float4*)&s_w2[rel * HH * CC + cc * HH]; // 64B-aligned
  float acc = 0.0f;
  #pragma unroll
  for (int q = 0; q < 4; ++q) {
    float4 hv = hn4[q];   // global_load_b128 (broadcast across the 8-lane group)
    float4 wv = wr4[q];   // ds_load_b128
    acc = fmaf(hv.x, wv.x, acc);
    acc = fmaf(hv.y, wv.y, acc);
    acc = fmaf(hv.z, wv.z, acc);
    acc = fmaf(hv.w, wv.w, acc);
  }

  float inv = 1.0f / fmaxf(cnt[dst * RR + rel], 1.0f);
  atomicAdd(&out[dst * CC + cc], acc * inv);
}

extern "C" void kernel_launch(void* const* d_in, const int* in_sizes, int n_in,
                              void* d_out, int out_size, void* d_ws, size_t ws_size,
                              hipStream_t stream) {
  const int*   ei     = (const int*)d_in[0];     // edge_index [2,E]
  const int*   et     = (const int*)d_in[1];     // edge_type  [E]
  const float* basis1 = (const float*)d_in[2];   // [B,N,H]
  const float* comp1  = (const float*)d_in[3];   // [R,B]
  const float* root1  = (const float*)d_in[4];   // [N,H]
  const float* bias1  = (const float*)d_in[5];   // [H]
  const float* basis2 = (const float*)d_in[6];   // [B,H,C]
  const float* comp2  = (const float*)d_in[7];   // [R,B]
  const float* root2  = (const float*)d_in[8];   // [H,C]
  const float* bias2  = (const float*)d_in[9];   // [C]
  (void)in_sizes; (void)n_in; (void)out_size; (void)ws_size;

  float* out  = (float*)d_out;
  float* ws   = (float*)d_ws;
  float* cnt  = ws + CNT_OFF;
  float* hbuf = ws + H_OFF;
  float* w2t  = ws + W2_OFF;

  k_zero   <<<(ZERO_TOTAL / 4 + 255) / 256, 256, 0, stream>>>((float4*)ws);
  k_count  <<<(EE + 255) / 256,           256, 0, stream>>>(ei, et, cnt);
  k_w2     <<<(RR * HH * CC + 255) / 256, 256, 0, stream>>>(comp2, basis2, w2t);
  k_layer1 <<<EE / 16,                    256, 0, stream>>>(ei, et, basis1, comp1, cnt, hbuf);
  k_hfinal <<<(NN * HH / 4 + 255) / 256,  256, 0, stream>>>((const float4*)root1, bias1,
                                                            (float4*)hbuf);
  k_root_wmma<<<NN / 32,                   64, 0, stream>>>(hbuf, root2, bias2, out);
  k_layer2 <<<EE / 32,                    256, 0, stream>>>(ei, et, hbuf, w2t, cnt, out);
}